// EppsPulley_20340965114056
// MI455X (gfx1250) — compile-verified
//
#include <hip/hip_runtime.h>
#include <math.h>

#define NPTS 17
#define NCH  34           // 17 cos channels then 17 sin channels

typedef __attribute__((ext_vector_type(2))) float v2f;
typedef __attribute__((ext_vector_type(8))) float v8f;

__device__ __forceinline__ float wave_sum(float v) {
#pragma unroll
    for (int off = 16; off > 0; off >>= 1)
        v += __shfl_xor(v, off, 32);
    return v;
}

// ---------------- Pass 1: sum & sumsq partials ----------------
__global__ void k_stats_partial(const float* __restrict__ x, int n,
                                float* __restrict__ part) {
    float s = 0.f, ss = 0.f;
    const int stride = gridDim.x * blockDim.x;
    const int gid = blockIdx.x * blockDim.x + threadIdx.x;
    const int n4 = n >> 2;
    const float4* x4 = (const float4*)x;
    for (int i = gid; i < n4; i += stride) {
        float4 v = x4[i];
        s += (v.x + v.y) + (v.z + v.w);
        ss = fmaf(v.x, v.x, fmaf(v.y, v.y, fmaf(v.z, v.z, fmaf(v.w, v.w, ss))));
    }
    for (int i = (n4 << 2) + gid; i < n; i += stride) {
        float v = x[i]; s += v; ss = fmaf(v, v, ss);
    }
    __shared__ float red[16];
    const int wave = threadIdx.x >> 5, lane = threadIdx.x & 31;
    float a = wave_sum(s), b = wave_sum(ss);
    if (lane == 0) { red[wave * 2] = a; red[wave * 2 + 1] = b; }
    __syncthreads();
    if (threadIdx.x == 0) {
        float S = 0.f, SS = 0.f;
        const int nw = blockDim.x >> 5;
        for (int w = 0; w < nw; ++w) { S += red[2 * w]; SS += red[2 * w + 1]; }
        part[blockIdx.x * 2]     = S;
        part[blockIdx.x * 2 + 1] = SS;
    }
}

// ---------------- Pass 2: finalize mu, 1/(sd+eps) ----------------
__global__ void k_stats_final(const float* __restrict__ part, int nblocks, int n,
                              float* __restrict__ stats) {
    float s = 0.f, ss = 0.f;
    for (int b = threadIdx.x; b < nblocks; b += blockDim.x) {
        s += part[2 * b]; ss += part[2 * b + 1];
    }
    __shared__ float red[16];
    const int wave = threadIdx.x >> 5, lane = threadIdx.x & 31;
    float a = wave_sum(s), b = wave_sum(ss);
    if (lane == 0) { red[wave * 2] = a; red[wave * 2 + 1] = b; }
    __syncthreads();
    if (threadIdx.x == 0) {
        float S = 0.f, SS = 0.f;
        const int nw = blockDim.x >> 5;
        for (int w = 0; w < nw; ++w) { S += red[2 * w]; SS += red[2 * w + 1]; }
        float fn  = (float)n;
        float mu  = S / fn;
        float var = (SS - S * S / fn) / (fn - 1.0f);   // ddof = 1
        var = var > 0.f ? var : 0.f;
        float sd  = sqrtf(var);
        stats[0] = mu;
        stats[1] = 1.0f / (sd + 1e-8f);
    }
}

// ---------------- Pass 3: ECF partials via Chebyshev recurrence ----------------
__device__ __forceinline__ void ecf_accum(float xv, float mu, float inv,
                                          float* __restrict__ acc) {
    const float dt = 2.0f / 17.0f;
    float th = (xv - mu) * inv * dt;
    float s1, c1;
    __sincosf(th, &s1, &c1);                 // one trans pair per element
    float tc = c1 + c1;
    float ck = c1, sk = s1, ckm = 1.f, skm = 0.f;
    acc[0]    += ck;
    acc[NPTS] += sk;
#pragma unroll
    for (int p = 1; p < NPTS; ++p) {         // cos((p+1)th), sin((p+1)th)
        float cn = fmaf(tc, ck, -ckm);
        float sn = fmaf(tc, sk, -skm);
        ckm = ck; skm = sk; ck = cn; sk = sn;
        acc[p]        += ck;
        acc[NPTS + p] += sk;
    }
}

__global__ void k_ecf_partial(const float* __restrict__ x, int n,
                              const float* __restrict__ stats,
                              float* __restrict__ part) {
    const float mu = stats[0], inv = stats[1];
    float acc[NCH];
#pragma unroll
    for (int c = 0; c < NCH; ++c) acc[c] = 0.f;

    const int stride = gridDim.x * blockDim.x;
    const int gid = blockIdx.x * blockDim.x + threadIdx.x;
    const int n4 = n >> 2;
    const float4* x4 = (const float4*)x;
    for (int i = gid; i < n4; i += stride) {
        float4 v = x4[i];
        ecf_accum(v.x, mu, inv, acc);
        ecf_accum(v.y, mu, inv, acc);
        ecf_accum(v.z, mu, inv, acc);
        ecf_accum(v.w, mu, inv, acc);
    }
    for (int i = (n4 << 2) + gid; i < n; i += stride)
        ecf_accum(x[i], mu, inv, acc);

    __shared__ float red[8 * NCH];
    const int wave = threadIdx.x >> 5, lane = threadIdx.x & 31;
#pragma unroll
    for (int c = 0; c < NCH; ++c) {
        float v = wave_sum(acc[c]);
        if (lane == 0) red[wave * NCH + c] = v;
    }
    __syncthreads();
    if (threadIdx.x < NCH) {
        float s = 0.f;
        const int nw = blockDim.x >> 5;
        for (int w = 0; w < nw; ++w) s += red[w * NCH + threadIdx.x];
        part[blockIdx.x * NCH + threadIdx.x] = s;
    }
}

// ---------------- Pass 4: WMMA cross-block reduce + finalize ----------------
// One wave (32 threads). Channels 0..31 summed over `nblocks` partials with
// V_WMMA_F32_16X16X4_F32 (B = ones => D[m][n] = sum_k A[m][k], fp32 exact).
// A layout (16x4 f32): lanes 0-15 hold K=0,1; lanes 16-31 hold K=2,3; M = lane&15.
__global__ void k_final(const float* __restrict__ part, int nblocks, int n,
                        float* __restrict__ out) {
    const int lane = threadIdx.x;
    const int half = lane >> 4;       // 0: K=0,1   1: K=2,3
    const int m    = lane & 15;

    v8f c0 = {};                       // channels 0..15
    v8f c1 = {};                       // channels 16..31
    v2f ones; ones.x = 1.0f; ones.y = 1.0f;

    for (int k0 = 0; k0 < nblocks; k0 += 4) {
        const float* p0 = part + (size_t)(k0 + (half ? 2 : 0)) * NCH;
        v2f a0; a0.x = p0[m];        a0.y = p0[NCH + m];         // channel m
        v2f a1; a1.x = p0[16 + m];   a1.y = p0[NCH + 16 + m];    // channel 16+m
        c0 = __builtin_amdgcn_wmma_f32_16x16x4_f32(false, a0, false, ones,
                                                   (short)0, c0, false, false);
        c1 = __builtin_amdgcn_wmma_f32_16x16x4_f32(false, a1, false, ones,
                                                   (short)0, c1, false, false);
    }

    // channels 32,33 reduced uniformly across the wave (EXEC stays all-1s above)
    float s32 = 0.f, s33 = 0.f;
    for (int b = lane; b < nblocks; b += 32) {
        s32 += part[(size_t)b * NCH + 32];
        s33 += part[(size_t)b * NCH + 33];
    }
    s32 = wave_sum(s32);
    s33 = wave_sum(s33);

    __shared__ float S[NCH];
    // D layout: VGPR r, lanes 0-15 -> M=r ; lanes 16-31 -> M=8+r (same value all N)
    if (lane == 0) {
#pragma unroll
        for (int r = 0; r < 8; ++r) { S[r] = c0[r]; S[16 + r] = c1[r]; }
        S[32] = s32; S[33] = s33;
    }
    if (lane == 16) {
#pragma unroll
        for (int r = 0; r < 8; ++r) { S[8 + r] = c0[r]; S[24 + r] = c1[r]; }
    }
    __syncthreads();

    if (lane == 0) {
        const float dt = 2.0f / 17.0f;
        const float invn = 1.0f / (float)n;
        float total = 0.f;
        for (int p = 0; p < NPTS; ++p) {
            float t   = dt * (float)(p + 1);
            float er  = S[p] * invn;
            float ei  = S[NPTS + p] * invn;
            float tcf = __expf(-0.5f * t * t);
            float integ = er * er + ei * ei - 2.0f * er * tcf + tcf * tcf;
            float w = (p == 0 || p == NPTS - 1) ? 0.5f * dt : dt;
            total = fmaf(w, integ, total);
        }
        out[0] = total;
    }
}

extern "C" void kernel_launch(void* const* d_in, const int* in_sizes, int n_in,
                              void* d_out, int out_size, void* d_ws, size_t ws_size,
                              hipStream_t stream) {
    const float* x = (const float*)d_in[0];
    const int n = in_sizes[0];
    float* ws = (float*)d_ws;

    int G = 1024;                                   // must be a multiple of 4
    size_t need = (size_t)(2 * G + 64 + G * NCH) * sizeof(float);
    if (ws_size < need) G = 128;                    // conservative fallback

    float* part1 = ws;                              // G * 2 floats
    float* stats = ws + 2 * G;                      // mu, inv_sd
    float* part3 = ws + 2 * G + 64;                 // G * NCH floats

    k_stats_partial<<<G, 256, 0, stream>>>(x, n, part1);
    k_stats_final <<<1, 256, 0, stream>>>(part1, G, n, stats);
    k_ecf_partial <<<G, 256, 0, stream>>>(x, n, stats, part3);
    k_final       <<<1, 32, 0, stream>>>(part3, G, n, (float*)d_out);
}